// Embedding_37864431682597
// MI455X (gfx1250) — compile-verified
//
#include <hip/hip_runtime.h>
#include <hip/hip_bf16.h>

// Embedding gather:  out[t, :] = table[x[t], :]
//   x:     8192 int32 tokens (B=4, C=2048)
//   table: 50257 x 768 float32
//   out:   8192 x 768 float32
//
// Pure bandwidth kernel: 48 MB total traffic -> ~2.1 us floor @ 23.3 TB/s.
// One workgroup (192 threads = 6 wave32) per token row; each lane moves one
// 16-byte vector (global_load_b128 / global_store_b128, 512 B coalesced per
// wave). Token id load is block-uniform -> scalar s_load path. Stores are
// non-temporal (write-once output) so the 24 MB result doesn't rinse table
// rows (repeated tokens) out of the near caches.

#define EMBED 768
#define VEC4_PER_ROW (EMBED / 4)   // 192

// Native clang vector type (HIP's float4 is a struct and is rejected by the
// nontemporal builtins).
typedef __attribute__((ext_vector_type(4))) float v4f;

__global__ __launch_bounds__(VEC4_PER_ROW)
void embed_gather_kernel(const int* __restrict__ x,
                         const float* __restrict__ table,
                         float* __restrict__ out,
                         int n_tok) {
    const int row = blockIdx.x;
    if (row >= n_tok) return;

    // Uniform (SGPR) load of the token id for this row.
    const int tok = x[row];

    const v4f* __restrict__ src =
        reinterpret_cast<const v4f*>(table) + (size_t)tok * VEC4_PER_ROW;
    v4f* __restrict__ dst =
        reinterpret_cast<v4f*>(out) + (size_t)row * VEC4_PER_ROW;

    // One 16-byte load + one 16-byte non-temporal store per lane.
    const v4f v = src[threadIdx.x];
    __builtin_nontemporal_store(v, &dst[threadIdx.x]);
}

extern "C" void kernel_launch(void* const* d_in, const int* in_sizes, int n_in,
                              void* d_out, int out_size, void* d_ws, size_t ws_size,
                              hipStream_t stream) {
    const int*   x     = (const int*)d_in[0];     // (B*C) int32 tokens
    const float* table = (const float*)d_in[1];   // (VOCAB, EMBED) f32
    float*       out   = (float*)d_out;           // (B*C, EMBED) f32

    const int n_tok = in_sizes[0];                // 8192

    dim3 grid(n_tok);
    dim3 block(VEC4_PER_ROW);                     // 192 threads = 6 waves
    hipLaunchKernelGGL(embed_gather_kernel, grid, block, 0, stream,
                       x, table, out, n_tok);
}